// GNNModel_35046933135499
// MI455X (gfx1250) — compile-verified
//
#include <hip/hip_runtime.h>
#include <math.h>

#define DIN  128
#define DHID 32
#define DOUTC 32
#define NEG_SLOPE 0.2f
#define EPSF 1e-16f

typedef float v2f __attribute__((ext_vector_type(2)));
typedef float v8f __attribute__((ext_vector_type(8)));

// order-preserving float<->int key (monotone under signed int compare)
__device__ __forceinline__ int fkey(float f) {
  int i = __float_as_int(f);
  return (i >= 0) ? i : (i ^ 0x7fffffff);
}
__device__ __forceinline__ float funkey(int k) {
  int i = (k >= 0) ? k : (k ^ 0x7fffffff);
  return __int_as_float(i);
}
__device__ __forceinline__ float lrelu(float v) { return v > 0.f ? v : NEG_SLOPE * v; }

// ---------------- init: zero accumulators, deg=1 (self-loop), esum=0 ----------------
__global__ void k_init(float* __restrict__ out, float* __restrict__ agg1,
                       float* __restrict__ deg, float* __restrict__ esum, int N) {
  int i = blockIdx.x * blockDim.x + threadIdx.x;
  if (i < N * 32) { out[i] = 0.f; agg1[i] = 0.f; }
  if (i < N)      { deg[i] = 1.0f; esum[i] = 0.f; }
}

// ---------------- degree accumulate over real edges ----------------
__global__ void k_deg(const long long* __restrict__ dst, float* __restrict__ deg, int E) {
  int e = blockIdx.x * blockDim.x + threadIdx.x;
  if (e < E) atomicAdd(&deg[dst[e]], 1.0f);
}

// ---------------- dis = rsqrt(deg), in place ----------------
__global__ void k_dis(float* __restrict__ deg, int N) {
  int i = blockIdx.x * blockDim.x + threadIdx.x;
  if (i < N) deg[i] = rsqrtf(deg[i]);
}

// ---------------- fp32 WMMA GEMM: C[M x 32] = A[M x K] * B[K x 32] ----------------
// One wave computes a 16x32 tile via two 16x16 accumulators, K stepped by 4
// using V_WMMA_F32_16X16X4_F32. 32-bit A frag: lanes 0-15 rows M, v0/v1 = K,K+1;
// lanes 16-31 = K+2,K+3. B frag mirrored with N in lanes. C/D: VGPR v -> row v + 8*half.
template <int K>
__global__ void k_gemm_wmma(const float* __restrict__ A, const float* __restrict__ B,
                            float* __restrict__ C, int M) {
  const int lane = threadIdx.x & 31;
  const int wave = (int)(((size_t)blockIdx.x * blockDim.x + threadIdx.x) >> 5);
  const int m0 = wave << 4;
  if (m0 >= M) return;                 // wave-uniform: EXEC stays all-ones for WMMA
  const int half = lane >> 4;
  const int r = lane & 15;
  v8f acc0 = {0.f, 0.f, 0.f, 0.f, 0.f, 0.f, 0.f, 0.f};
  v8f acc1 = acc0;
  const float* arow = A + (size_t)(m0 + r) * K + 2 * half;
#pragma unroll
  for (int k = 0; k < K; k += 4) {
    v2f a;
    a.x = arow[k];
    a.y = arow[k + 1];
    const float* bp = B + (size_t)(k + 2 * half) * 32;
    v2f blo, bhi;
    blo.x = bp[r];       blo.y = bp[32 + r];        // N = r,    K = k+2h, k+2h+1
    bhi.x = bp[16 + r];  bhi.y = bp[48 + r];        // N = r+16
    acc0 = __builtin_amdgcn_wmma_f32_16x16x4_f32(false, a, false, blo, (short)0, acc0, false, false);
    acc1 = __builtin_amdgcn_wmma_f32_16x16x4_f32(false, a, false, bhi, (short)0, acc1, false, false);
  }
  float* crow = C + (size_t)(m0 + half * 8) * 32;
#pragma unroll
  for (int v = 0; v < 8; ++v) {
    crow[(size_t)v * 32 + r]      = acc0[v];
    crow[(size_t)v * 32 + 16 + r] = acc1[v];
  }
}

// ---------------- GCN edge aggregation: wave per edge, lane = channel ----------------
__global__ void k_gcn_edge(const long long* __restrict__ src, const long long* __restrict__ dst,
                           const float* __restrict__ dis, const float* __restrict__ h,
                           float* __restrict__ agg, int E) {
  int wid = (int)(((size_t)blockIdx.x * blockDim.x + threadIdx.x) >> 5);
  int lane = threadIdx.x & 31;
  if (wid >= E) return;
  long long s = src[wid], d = dst[wid];
  float nrm = dis[s] * dis[d];
  atomicAdd(&agg[d * 32 + lane], nrm * h[s * 32 + lane]);
}

// ---------------- GCN finalize (+self-loop +bias, ReLU), in place into agg1 ----------------
__global__ void k_gcn_fin(float* __restrict__ agg1, const float* __restrict__ h,
                          const float* __restrict__ dis, const float* __restrict__ b1, int N) {
  int i = blockIdx.x * blockDim.x + threadIdx.x;
  if (i >= N * 32) return;
  int node = i >> 5, c = i & 31;
  float v = agg1[i] + dis[node] * dis[node] * h[i] + b1[c];
  agg1[i] = v > 0.f ? v : 0.f;
}

// ---------------- attention scalars per node (wave per node), init emax with self-loop ----------------
__global__ void k_att(const float* __restrict__ g, const float* __restrict__ att_src,
                      const float* __restrict__ att_dst, float* __restrict__ a_s,
                      float* __restrict__ a_d, int* __restrict__ emax_key, int N) {
  int wid = (int)(((size_t)blockIdx.x * blockDim.x + threadIdx.x) >> 5);
  int lane = threadIdx.x & 31;
  if (wid >= N) return;
  float gv = g[(size_t)wid * 32 + lane];
  float s1 = gv * att_src[lane];
  float s2 = gv * att_dst[lane];
#pragma unroll
  for (int off = 16; off > 0; off >>= 1) {
    s1 += __shfl_xor(s1, off, 32);
    s2 += __shfl_xor(s2, off, 32);
  }
  if (lane == 0) {
    a_s[wid] = s1;
    a_d[wid] = s2;
    emax_key[wid] = fkey(lrelu(s1 + s2));   // self-loop edge seeds the segment max
  }
}

// ---------------- GAT segment-max over real edges (scalar, thread per edge) ----------------
__global__ void k_gat_max(const long long* __restrict__ src, const long long* __restrict__ dst,
                          const float* __restrict__ a_s, const float* __restrict__ a_d,
                          int* __restrict__ emax_key, int E) {
  int e = blockIdx.x * blockDim.x + threadIdx.x;
  if (e >= E) return;
  long long s = src[e], d = dst[e];
  atomicMax(&emax_key[d], fkey(lrelu(a_s[s] + a_d[d])));
}

// ---------------- fused GAT exp-sum + weighted aggregation (wave per edge) ----------------
__global__ void k_gat_agg(const long long* __restrict__ src, const long long* __restrict__ dst,
                          const float* __restrict__ a_s, const float* __restrict__ a_d,
                          const int* __restrict__ emax_key, const float* __restrict__ g,
                          float* __restrict__ out, float* __restrict__ esum, int E) {
  int wid = (int)(((size_t)blockIdx.x * blockDim.x + threadIdx.x) >> 5);
  int lane = threadIdx.x & 31;
  if (wid >= E) return;
  long long s = src[wid], d = dst[wid];
  float v = lrelu(a_s[s] + a_d[d]);
  float ee = __expf(v - funkey(emax_key[d]));
  atomicAdd(&out[d * 32 + lane], ee * g[s * 32 + lane]);
  if (lane == 0) atomicAdd(&esum[d], ee);
}

// ---------------- GAT finalize: add self-loop term, normalize, +bias ----------------
__global__ void k_fin(float* __restrict__ out, const float* __restrict__ g,
                      const float* __restrict__ a_s, const float* __restrict__ a_d,
                      const int* __restrict__ emax_key, const float* __restrict__ esum,
                      const float* __restrict__ b2, int N) {
  int i = blockIdx.x * blockDim.x + threadIdx.x;
  if (i >= N * 32) return;
  int node = i >> 5, c = i & 31;
  float e = lrelu(a_s[node] + a_d[node]);
  float ee = __expf(e - funkey(emax_key[node]));
  out[i] = (out[i] + ee * g[i]) / (esum[node] + ee + EPSF) + b2[c];
}

extern "C" void kernel_launch(void* const* d_in, const int* in_sizes, int n_in,
                              void* d_out, int out_size, void* d_ws, size_t ws_size,
                              hipStream_t stream) {
  const float*     x       = (const float*)d_in[0];
  const long long* ei      = (const long long*)d_in[1];
  const float*     W1      = (const float*)d_in[2];
  const float*     b1      = (const float*)d_in[3];
  const float*     W2      = (const float*)d_in[4];
  const float*     att_src = (const float*)d_in[5];
  const float*     att_dst = (const float*)d_in[6];
  const float*     b2      = (const float*)d_in[7];

  const int N = in_sizes[0] / DIN;   // 100000
  const int E = in_sizes[1] / 2;     // 1600000
  const long long* srcp = ei;
  const long long* dstp = ei + E;

  // workspace layout (floats): dis | h | agg1(->hrelu) | g | a_s | a_d | esum | emax_key(int)
  float* ws    = (float*)d_ws;
  float* dis   = ws;                       // N  (holds deg, then rsqrt in place)
  float* h     = dis + N;                  // N*32
  float* agg1  = h + (size_t)N * 32;       // N*32
  float* g     = agg1 + (size_t)N * 32;    // N*32
  float* a_s   = g + (size_t)N * 32;       // N
  float* a_d   = a_s + N;                  // N
  float* esum  = a_d + N;                  // N
  int*   emaxk = (int*)(esum + N);         // N
  float* out   = (float*)d_out;            // N*32, used as GAT accumulator in place

  const int B = 256;
  const int gN32 = (N * 32 + B - 1) / B;
  const int gN   = (N + B - 1) / B;
  const int gE   = (E + B - 1) / B;
  const int gE32 = (int)(((size_t)E * 32 + B - 1) / B);
  const int mtiles = (N + 15) / 16;
  const int gGemm = (mtiles + 3) / 4;      // 128 threads = 4 waves per block

  k_init<<<gN32, B, 0, stream>>>(out, agg1, dis, esum, N);
  k_deg<<<gE, B, 0, stream>>>(dstp, dis, E);
  k_dis<<<gN, B, 0, stream>>>(dis, N);
  k_gemm_wmma<DIN><<<gGemm, 128, 0, stream>>>(x, W1, h, N);            // h = x @ W1
  k_gcn_edge<<<gE32, B, 0, stream>>>(srcp, dstp, dis, h, agg1, E);
  k_gcn_fin<<<gN32, B, 0, stream>>>(agg1, h, dis, b1, N);              // agg1 = relu(GCN out)
  k_gemm_wmma<DHID><<<gGemm, 128, 0, stream>>>(agg1, W2, g, N);        // g = hrelu @ W2
  k_att<<<gN32, B, 0, stream>>>(g, att_src, att_dst, a_s, a_d, emaxk, N);
  k_gat_max<<<gE, B, 0, stream>>>(srcp, dstp, a_s, a_d, emaxk, E);
  k_gat_agg<<<gE32, B, 0, stream>>>(srcp, dstp, a_s, a_d, emaxk, g, out, esum, E);
  k_fin<<<gN32, B, 0, stream>>>(out, g, a_s, a_d, emaxk, esum, b2, N);
}